// GaussianModel_24979529793553
// MI455X (gfx1250) — compile-verified
//
#include <hip/hip_runtime.h>

// Per-element covariance math.
// Reference: q = r/||r||; R(q); L = R * diag(exp(s)); cov = L L^T (upper tri).
// We use k = 2/||r||^2 on raw components (R is homogeneous degree-2 in the
// normalized quaternion) -> algebraically identical, avoids rsqrt + 4 muls.
__device__ __forceinline__ void cov6(const float4 r,
                                     float s0, float s1, float s2,
                                     float* __restrict__ o) {
    const float w = r.x, x = r.y, y = r.z, z = r.w;
    const float n2 = w * w + x * x + y * y + z * z;
    const float k  = 2.0f / n2;

    const float sx = __expf(s0);   // v_exp_f32
    const float sy = __expf(s1);
    const float sz = __expf(s2);

    const float xx = k * x * x, yy = k * y * y, zz = k * z * z;
    const float xy = k * x * y, xz = k * x * z, yz = k * y * z;
    const float wx = k * w * x, wy = k * w * y, wz = k * w * z;

    const float R00 = 1.0f - (yy + zz), R01 = xy - wz,          R02 = xz + wy;
    const float R10 = xy + wz,          R11 = 1.0f - (xx + zz), R12 = yz - wx;
    const float R20 = xz - wy,          R21 = yz + wx,          R22 = 1.0f - (xx + yy);

    const float L00 = R00 * sx, L01 = R01 * sy, L02 = R02 * sz;
    const float L10 = R10 * sx, L11 = R11 * sy, L12 = R12 * sz;
    const float L20 = R20 * sx, L21 = R21 * sy, L22 = R22 * sz;

    o[0] = L00 * L00 + L01 * L01 + L02 * L02;   // c00
    o[1] = L00 * L10 + L01 * L11 + L02 * L12;   // c01
    o[2] = L00 * L20 + L01 * L21 + L02 * L22;   // c02
    o[3] = L10 * L10 + L11 * L11 + L12 * L12;   // c11
    o[4] = L10 * L20 + L11 * L21 + L12 * L22;   // c12
    o[5] = L20 * L20 + L21 * L21 + L22 * L22;   // c22
}

// Streaming bandwidth kernel: 4 Gaussians per thread-iteration so EVERY
// access is a 16B-aligned 128-bit op:
//   scaling : 48B  = 3x global_load_b128
//   rotation: 64B  = 4x global_load_b128
//   output  : 96B  = 6x global_store_b128
// All index math in 32-bit (max buffer 96MB), widened only at the pointer add.
__global__ void cov3d_kernel(const float* __restrict__ scl,
                             const float4* __restrict__ rot,
                             float* __restrict__ out,
                             int n) {
    const int Q      = n >> 2;                         // element quads
    const int stride = (int)(gridDim.x * blockDim.x);
    int q = (int)(blockIdx.x * blockDim.x + threadIdx.x);

    for (; q < Q; q += stride) {
        // CDNA5: prefetch next grid-stride iteration (global_prefetch_b8)
        const int qn = q + stride;
        if (qn < Q) {
            __builtin_prefetch(rot + 4 * (long long)qn, 0, 3);
            __builtin_prefetch(scl + 12 * (long long)qn, 0, 3);
        }

        const float4* s4 = (const float4*)(scl + 12 * q);  // 48q bytes: 16B aligned
        const float4 sA = s4[0];   // s[e0].xyz, s[e1].x
        const float4 sB = s4[1];   // s[e1].yz,  s[e2].xy
        const float4 sC = s4[2];   // s[e2].z,   s[e3].xyz

        const float4 r0 = rot[4 * q + 0];
        const float4 r1 = rot[4 * q + 1];
        const float4 r2 = rot[4 * q + 2];
        const float4 r3 = rot[4 * q + 3];

        float o[24];
        cov6(r0, sA.x, sA.y, sA.z, o + 0);
        cov6(r1, sA.w, sB.x, sB.y, o + 6);
        cov6(r2, sB.z, sB.w, sC.x, o + 12);
        cov6(r3, sC.y, sC.z, sC.w, o + 18);

        float4* o4 = (float4*)(out + 24 * q);              // 96q bytes: 16B aligned
        #pragma unroll
        for (int j = 0; j < 6; ++j)
            o4[j] = make_float4(o[4 * j], o[4 * j + 1], o[4 * j + 2], o[4 * j + 3]);
    }

    // n%4 tail (not hit for N=4,000,000, kept for generality)
    const int rem  = n & 3;
    const int tid  = (int)(blockIdx.x * blockDim.x + threadIdx.x);
    if (tid < rem) {
        const int i = n - rem + tid;
        const float4 r = rot[i];
        float o[6];
        cov6(r, scl[3 * i], scl[3 * i + 1], scl[3 * i + 2], o);
        #pragma unroll
        for (int j = 0; j < 6; ++j) out[6 * i + j] = o[j];
    }
}

extern "C" void kernel_launch(void* const* d_in, const int* in_sizes, int n_in,
                              void* d_out, int out_size, void* d_ws, size_t ws_size,
                              hipStream_t stream) {
    (void)n_in; (void)d_ws; (void)ws_size; (void)out_size;

    const float*  scl = (const float*)d_in[0];    // (N,3) fp32
    const float4* rot = (const float4*)d_in[1];   // (N,4) fp32, 16B aligned
    float*        out = (float*)d_out;            // (N,6) fp32

    const int n = in_sizes[0] / 3;                // N = 4,000,000
    const int quads = n >> 2;                     // 1M quads

    // 1024 blocks x 256 threads = 256K threads = 8K wave32 waves, ~4 quads per
    // thread: enough MLP to saturate 23.3 TB/s and keeps the one-iteration-
    // ahead prefetch meaningful.
    int blocks = (quads + 255) / 256;
    if (blocks > 1024) blocks = 1024;
    if (blocks < 1)    blocks = 1;

    cov3d_kernel<<<blocks, 256, 0, stream>>>(scl, rot, out, n);
}